// GNN_4836133175913
// MI455X (gfx1250) — compile-verified
//
#include <hip/hip_runtime.h>
#include <hip/hip_bf16.h>

typedef float v2f __attribute__((ext_vector_type(2)));
typedef float v8f __attribute__((ext_vector_type(8)));

// ---------------------------------------------------------------------------
// order-preserving float<->uint key for atomic segment-max
// ---------------------------------------------------------------------------
__device__ __forceinline__ unsigned fkey(float f) {
    int i = __float_as_int(f);
    return ((unsigned)i) ^ ((unsigned)(i >> 31) | 0x80000000u);
}
__device__ __forceinline__ float funkey(unsigned u) {
    int i = (u & 0x80000000u) ? (int)(u ^ 0x80000000u) : (int)(~u);
    return __int_as_float(i);
}

// ---------------------------------------------------------------------------
// zero a float region (agg + mkey + s are laid out contiguously)
// ---------------------------------------------------------------------------
__global__ void zero_kernel(float* __restrict__ p, long long n) {
    long long t = (long long)blockIdx.x * blockDim.x + threadIdx.x;
    if (t < n) p[t] = 0.0f;
}

// ---------------------------------------------------------------------------
// Dual GEMM: Yl = X@Wl + bl, Yr = X@Wr + br.  X:[N,K], W:[K,64], Y:[N,64]
// block = 128 threads (4 waves); wave w -> column tile 16*w; blockIdx.x -> 16 rows.
// fp32 WMMA 16x16x4, A fragment shared across both accumulators.
// Requires N % 16 == 0 (true: 50000 = 3125*16) and K % 4 == 0.
// ---------------------------------------------------------------------------
__global__ void gemm16x16_dual(const float* __restrict__ X, int K,
                               const float* __restrict__ Wl, const float* __restrict__ bl,
                               const float* __restrict__ Wr, const float* __restrict__ br,
                               float* __restrict__ Yl, float* __restrict__ Yr) {
    const int lane = threadIdx.x & 31;
    const int wave = threadIdx.x >> 5;
    const int row0 = blockIdx.x * 16;
    const int col0 = wave * 16;
    const int lr   = lane & 15;
    const int half = lane >> 4;

    const float* xrow = X + (size_t)(row0 + lr) * K;
    v8f cl = {}; v8f cr = {};
    for (int k = 0; k < K; k += 4) {
        const int kk = k + 2 * half;
        v2f a;  a.x  = xrow[kk];                         a.y  = xrow[kk + 1];
        v2f bL; bL.x = Wl[(size_t)kk * 64 + col0 + lr];  bL.y = Wl[(size_t)(kk + 1) * 64 + col0 + lr];
        v2f bR; bR.x = Wr[(size_t)kk * 64 + col0 + lr];  bR.y = Wr[(size_t)(kk + 1) * 64 + col0 + lr];
        cl = __builtin_amdgcn_wmma_f32_16x16x4_f32(false, a, false, bL, (short)0, cl, false, false);
        cr = __builtin_amdgcn_wmma_f32_16x16x4_f32(false, a, false, bR, (short)0, cr, false, false);
    }
    const float biasl = bl[col0 + lr];
    const float biasr = br[col0 + lr];
#pragma unroll
    for (int j = 0; j < 8; ++j) {
        const size_t o = (size_t)(row0 + j + 8 * half) * 64 + col0 + lr;
        Yl[o] = cl[j] + biasl;
        Yr[o] = cr[j] + biasr;
    }
}

// ---------------------------------------------------------------------------
// Single GEMM with fused activation: Y = act(X@W + b).  act: 1=relu, 2=sigmoid
// ---------------------------------------------------------------------------
__global__ void gemm16x16_act(const float* __restrict__ X, int K,
                              const float* __restrict__ W, const float* __restrict__ b,
                              float* __restrict__ Y, int act) {
    const int lane = threadIdx.x & 31;
    const int wave = threadIdx.x >> 5;
    const int row0 = blockIdx.x * 16;
    const int col0 = wave * 16;
    const int lr   = lane & 15;
    const int half = lane >> 4;

    const float* xrow = X + (size_t)(row0 + lr) * K;
    v8f c = {};
    for (int k = 0; k < K; k += 4) {
        const int kk = k + 2 * half;
        v2f a;  a.x  = xrow[kk];                        a.y  = xrow[kk + 1];
        v2f bf; bf.x = W[(size_t)kk * 64 + col0 + lr];  bf.y = W[(size_t)(kk + 1) * 64 + col0 + lr];
        c = __builtin_amdgcn_wmma_f32_16x16x4_f32(false, a, false, bf, (short)0, c, false, false);
    }
    const float bias = b[col0 + lr];
#pragma unroll
    for (int j = 0; j < 8; ++j) {
        float v = c[j] + bias;
        if (act == 1) v = fmaxf(v, 0.0f);
        else if (act == 2) v = 1.0f / (1.0f + __expf(-v));
        Y[(size_t)(row0 + j + 8 * half) * 64 + col0 + lr] = v;
    }
}

// ---------------------------------------------------------------------------
// Edge pass A: logits + atomic segment-max (per dst,head)
// one thread per (edge, head)
// ---------------------------------------------------------------------------
__global__ void edge_logit(const float* __restrict__ xl, const float* __restrict__ xr,
                           const int* __restrict__ ei, const float* __restrict__ att,
                           float* __restrict__ logits, unsigned* __restrict__ mkey,
                           int Eorig, int Etot) {
    int t = blockIdx.x * blockDim.x + threadIdx.x;
    if (t >= Etot * 4) return;
    const int e = t >> 2, h = t & 3;
    int src, dst;
    if (e < Eorig) { src = ei[e]; dst = ei[Eorig + e]; }
    else           { src = dst = e - Eorig; }

    const float4* pl = (const float4*)(xl + (size_t)src * 64 + h * 16);
    const float4* pr = (const float4*)(xr + (size_t)dst * 64 + h * 16);
    const float4* pa = (const float4*)(att + h * 16);
    float logit = 0.0f;
#pragma unroll
    for (int i = 0; i < 4; ++i) {
        const float4 a = pl[i], b = pr[i], w = pa[i];
        float vx = a.x + b.x, vy = a.y + b.y, vz = a.z + b.z, vw = a.w + b.w;
        vx = vx > 0.0f ? vx : 0.2f * vx;
        vy = vy > 0.0f ? vy : 0.2f * vy;
        vz = vz > 0.0f ? vz : 0.2f * vz;
        vw = vw > 0.0f ? vw : 0.2f * vw;
        logit += w.x * vx + w.y * vy + w.z * vz + w.w * vw;
    }
    logits[t] = logit;
    atomicMax(mkey + (size_t)dst * 4 + h, fkey(logit));
}

// ---------------------------------------------------------------------------
// Edge pass B: p = exp(logit - m[dst]); atomic segment-sum
// ---------------------------------------------------------------------------
__global__ void edge_expsum(float* __restrict__ logits, const int* __restrict__ ei,
                            const unsigned* __restrict__ mkey, float* __restrict__ s,
                            int Eorig, int Etot) {
    int t = blockIdx.x * blockDim.x + threadIdx.x;
    if (t >= Etot * 4) return;
    const int e = t >> 2, h = t & 3;
    const int dst = (e < Eorig) ? ei[Eorig + e] : (e - Eorig);
    const float m = funkey(mkey[(size_t)dst * 4 + h]);
    const float p = __expf(logits[t] - m);
    logits[t] = p;
    atomicAdd(s + (size_t)dst * 4 + h, p);
}

// ---------------------------------------------------------------------------
// Edge pass C: alpha = p / s[dst]; agg[dst] += alpha * xl[src]  (16 ch/head)
// ---------------------------------------------------------------------------
__global__ void edge_scatter(const float* __restrict__ logits, const int* __restrict__ ei,
                             const float* __restrict__ s, const float* __restrict__ xl,
                             float* __restrict__ agg, int Eorig, int Etot) {
    int t = blockIdx.x * blockDim.x + threadIdx.x;
    if (t >= Etot * 4) return;
    const int e = t >> 2, h = t & 3;
    int src, dst;
    if (e < Eorig) { src = ei[e]; dst = ei[Eorig + e]; }
    else           { src = dst = e - Eorig; }
    const float alpha = logits[t] / s[(size_t)dst * 4 + h];
    const float4* pl = (const float4*)(xl + (size_t)src * 64 + h * 16);
    float* out = agg + (size_t)dst * 64 + h * 16;
#pragma unroll
    for (int i = 0; i < 4; ++i) {
        const float4 v = pl[i];
        atomicAdd(out + 4 * i + 0, alpha * v.x);
        atomicAdd(out + 4 * i + 1, alpha * v.y);
        atomicAdd(out + 4 * i + 2, alpha * v.z);
        atomicAdd(out + 4 * i + 3, alpha * v.w);
    }
}

// ---------------------------------------------------------------------------
// Node finalize kernels
// ---------------------------------------------------------------------------
__global__ void final_relu(const float* __restrict__ agg, const float* __restrict__ bias,
                           float* __restrict__ y, long long n) {
    long long t = (long long)blockIdx.x * blockDim.x + threadIdx.x;
    if (t < n) y[t] = fmaxf(agg[t] + bias[t & 63], 0.0f);
}

__global__ void final_mix(const float* __restrict__ agg, const float* __restrict__ bias,
                          const float* __restrict__ x1, float* __restrict__ y, long long n) {
    long long t = (long long)blockIdx.x * blockDim.x + threadIdx.x;
    if (t < n) y[t] = 0.5f * fmaxf(agg[t] + bias[t & 63], 0.0f) + 0.5f * x1[t];
}

// ---------------------------------------------------------------------------
// host side
// ---------------------------------------------------------------------------
extern "C" void kernel_launch(void* const* d_in, const int* in_sizes, int n_in,
                              void* d_out, int out_size, void* d_ws, size_t ws_size,
                              hipStream_t stream) {
    const float* x    = (const float*)d_in[0];
    const int*   ei   = (const int*)d_in[1];
    const float* Wl0  = (const float*)d_in[2];
    const float* bl0  = (const float*)d_in[3];
    const float* Wr0  = (const float*)d_in[4];
    const float* br0  = (const float*)d_in[5];
    const float* att0 = (const float*)d_in[6];
    const float* b0   = (const float*)d_in[7];
    const float* Wl1  = (const float*)d_in[8];
    const float* bl1  = (const float*)d_in[9];
    const float* Wr1  = (const float*)d_in[10];
    const float* br1  = (const float*)d_in[11];
    const float* att1 = (const float*)d_in[12];
    const float* b1   = (const float*)d_in[13];
    const float* W1   = (const float*)d_in[14];
    const float* bias1= (const float*)d_in[15];
    const float* W2   = (const float*)d_in[16];
    const float* bias2= (const float*)d_in[17];
    float* out = (float*)d_out;

    const int N     = in_sizes[0] / 128;
    const int Eorig = in_sizes[1] / 2;
    const int Etot  = Eorig + N;

    float* ws = (float*)d_ws;
    const size_t NH  = (size_t)N * 64;
    const size_t NH4 = (size_t)N * 4;
    float*    xl     = ws;                     // [N,64]  (reused as x_last later)
    float*    xr     = ws + NH;                // [N,64]  (reused as h later)
    float*    agg    = ws + 2 * NH;            // [N,64]
    unsigned* mkey   = (unsigned*)(ws + 3 * NH);           // [N,4]
    float*    s      = ws + 3 * NH + NH4;                  // [N,4]
    float*    x1     = ws + 3 * NH + 2 * NH4;              // [N,64]
    float*    logits = ws + 4 * NH + 2 * NH4;              // [Etot,4]
    float*    xlast  = xl;
    float*    hbuf   = xr;

    const int T = 256;
    const long long zn = (long long)(NH + 2 * NH4);
    const unsigned zb  = (unsigned)((zn + T - 1) / T);
    const unsigned eb  = (unsigned)(((long long)Etot * 4 + T - 1) / T);
    const unsigned nb  = (unsigned)(((long long)NH + T - 1) / T);
    const unsigned gm  = (unsigned)(N / 16);   // N = 50000 = 3125 * 16

    // ---- layer 0 ----
    zero_kernel<<<zb, T, 0, stream>>>(agg, zn);
    gemm16x16_dual<<<gm, 128, 0, stream>>>(x, 128, Wl0, bl0, Wr0, br0, xl, xr);
    edge_logit  <<<eb, T, 0, stream>>>(xl, xr, ei, att0, logits, mkey, Eorig, Etot);
    edge_expsum <<<eb, T, 0, stream>>>(logits, ei, mkey, s, Eorig, Etot);
    edge_scatter<<<eb, T, 0, stream>>>(logits, ei, s, xl, agg, Eorig, Etot);
    final_relu  <<<nb, T, 0, stream>>>(agg, b0, x1, (long long)NH);

    // ---- layer 1 ----
    zero_kernel<<<zb, T, 0, stream>>>(agg, zn);
    gemm16x16_dual<<<gm, 128, 0, stream>>>(x1, 64, Wl1, bl1, Wr1, br1, xl, xr);
    edge_logit  <<<eb, T, 0, stream>>>(xl, xr, ei, att1, logits, mkey, Eorig, Etot);
    edge_expsum <<<eb, T, 0, stream>>>(logits, ei, mkey, s, Eorig, Etot);
    edge_scatter<<<eb, T, 0, stream>>>(logits, ei, s, xl, agg, Eorig, Etot);
    final_mix   <<<nb, T, 0, stream>>>(agg, b1, x1, xlast, (long long)NH);

    // ---- MLP head ----
    gemm16x16_act<<<gm, 128, 0, stream>>>(xlast, 64, W1, bias1, hbuf, 1);
    gemm16x16_act<<<gm, 128, 0, stream>>>(hbuf, 64, W2, bias2, out, 2);
}